// RotateChordCircularPad_86277303042520
// MI455X (gfx1250) — compile-verified
//
#include <hip/hip_runtime.h>
#include <hip/hip_bf16.h>

// ---- problem constants (from the reference: C = 240 = 15 tracks * 16 floats) ----
#define TS        16      // floats per track chunk
#define TRACKS    15
#define CDIM      240
#define LANES     32      // wave32
#define WAVES_PB  4       // waves per block (blockDim = 128)
#define NSTEPS    16      // pipeline steps per wave
#define CHUNK_B   64      // bytes per lane-chunk (16 floats)
#define STEP_B    (LANES * CHUNK_B)   // 2048 B per wave-step
#define BUFS      4       // LDS ring depth (proven hazard-free with wait<=4)

// 4 async B128 loads: global -> LDS. INST_OFFSET applies to BOTH the LDS and
// the global address (CDNA5 ISA pseudocode), so one LDS base VGPR
// + one 64-bit global address VGPR pair covers the whole 64-byte chunk.
// Loads keep default (RT) policy: x fits in the 192MB L2 and is re-read.
__device__ __forceinline__ void async_load_chunk(unsigned lds_addr, unsigned long long gsrc) {
    asm volatile(
        "global_load_async_to_lds_b128 %0, %1, off\n\t"
        "global_load_async_to_lds_b128 %0, %1, off offset:16\n\t"
        "global_load_async_to_lds_b128 %0, %1, off offset:32\n\t"
        "global_load_async_to_lds_b128 %0, %1, off offset:48"
        :: "v"(lds_addr), "v"(gsrc) : "memory");
}

// 4 async B128 stores: LDS -> global. Output is write-once / never re-read:
// non-temporal so the 126MB output stream does not evict the L2-resident input.
__device__ __forceinline__ void async_store_chunk(unsigned long long gdst, unsigned lds_addr) {
    asm volatile(
        "global_store_async_from_lds_b128 %0, %1, off th:TH_STORE_NT\n\t"
        "global_store_async_from_lds_b128 %0, %1, off offset:16 th:TH_STORE_NT\n\t"
        "global_store_async_from_lds_b128 %0, %1, off offset:32 th:TH_STORE_NT\n\t"
        "global_store_async_from_lds_b128 %0, %1, off offset:48 th:TH_STORE_NT"
        :: "v"(gdst), "v"(lds_addr) : "memory");
}

// Addresses for chunk (j, t) of one batch. shift_t = 2^(t-1) (0 for t==0)
// == (1<<t)>>1. j+shift < padded_len here (min(len)>N/2), loop is general.
__device__ __forceinline__ void calc_addrs(
    unsigned jj, unsigned t, int padded, int len,
    const float* __restrict__ xb, float* __restrict__ outb,
    unsigned long long& gsrc, unsigned long long& gdst)
{
    int shift = (int)((1u << t) >> 1);
    int s     = (int)jj + shift;
    while (s >= padded) s -= padded;            // here: 0 iterations
    int row = s;
    while (row >= len) row -= len;              // here: <= 2 iterations
    gsrc = (unsigned long long)(xb   + (unsigned long long)(unsigned)row * CDIM + t * TS);
    gdst = (unsigned long long)(outb + (unsigned long long)jj            * CDIM + t * TS);
}

__global__ __launch_bounds__(WAVES_PB * LANES)
void rotate_chord_gather_async(const float* __restrict__ x,
                               const int*   __restrict__ lengths,
                               float*       __restrict__ out,
                               int B, int N, int max_len,
                               unsigned cpb /* = max_len * TRACKS */)
{
    __shared__ __align__(16) unsigned char smem[WAVES_PB * BUFS * STEP_B]; // 32 KB
    const unsigned lane = threadIdx.x & (LANES - 1);
    const unsigned wave = threadIdx.x >> 5;
    const int      b    = blockIdx.y;          // one batch per grid row

    // padded_len = 3 * min(lengths). Tiny loop, keep it scalar.
    int minlen = lengths[0];
    #pragma clang loop vectorize(disable) unroll(disable)
    for (int i = 1; i < B; ++i) { int l = lengths[i]; minlen = (l < minlen) ? l : minlen; }
    const int padded = 3 * minlen;
    const int len    = lengths[b];             // block-uniform -> scalar load

    const unsigned wave_id = blockIdx.x * WAVES_PB + wave;
    const unsigned q0 = wave_id * (LANES * NSTEPS);
    if (q0 >= cpb) return;                     // uniform within a wave

    const float* xb   = x   + (unsigned long long)b * (unsigned)N       * CDIM;
    float*       outb = out + (unsigned long long)b * (unsigned)max_len * CDIM;

    // addrspace(3) offset: low 32 bits of the generic (flat) pointer.
    const unsigned lds_wave = (unsigned)(unsigned long long)(const void*)&smem[0]
                              + wave * (BUFS * STEP_B);
    const unsigned lds_lane = lds_wave + lane * CHUNK_B;

    // (j, t) for this lane's step-0 chunk; one division per wave, then
    // incremental update: q += 32 == 2 rows + 2 tracks.
    const unsigned qs   = q0 + lane;
    unsigned j = qs / TRACKS;                  // constant divisor -> mulhi
    unsigned t = qs - j * TRACKS;
    const unsigned maxj = (unsigned)(max_len - 1);

    unsigned long long g_ld, g_st_cur, g_st_next = 0;

    // prologue: loads for step 0 into buf 0
    calc_addrs((j < maxj ? j : maxj), t, padded, len, xb, outb, g_ld, g_st_cur);
    async_load_chunk(lds_lane, g_ld);

    #pragma unroll
    for (int k = 0; k < NSTEPS; ++k) {
        if (k + 1 < NSTEPS) {
            t += 2u; j += 2u;
            if (t >= TRACKS) { t -= TRACKS; ++j; }
            calc_addrs((j < maxj ? j : maxj), t, padded, len, xb, outb, g_ld, g_st_next);
            async_load_chunk(lds_lane + ((unsigned)(k + 1) & (BUFS - 1)) * STEP_B, g_ld);
            // cnt<=4: loads(k) complete (else loads(k+1) would push cnt>=5) and
            // stores(k-2) complete -> buf[k&3] safe to read; ring distance 4 safe to write.
            asm volatile("s_wait_asynccnt 0x4" ::: "memory");
        } else {
            asm volatile("s_wait_asynccnt 0x0" ::: "memory");
        }
        async_store_chunk(g_st_cur, lds_lane + ((unsigned)k & (BUFS - 1)) * STEP_B);
        g_st_cur = g_st_next;
    }
    // s_endpgm performs an implicit wait-idle, flushing the final stores.
}

extern "C" void kernel_launch(void* const* d_in, const int* in_sizes, int n_in,
                              void* d_out, int out_size, void* d_ws, size_t ws_size,
                              hipStream_t stream) {
    const float* x       = (const float*)d_in[0];
    const int*   lengths = (const int*)d_in[1];
    float*       out     = (float*)d_out;

    const int B = in_sizes[1];                                   // 8
    const long long total_x = (long long)in_sizes[0];            // B*N*C
    const int N       = (int)(total_x / ((long long)B * CDIM));  // 16384
    const int max_len = out_size / (B * CDIM);                   // 16384

    const unsigned cpb = (unsigned)max_len * TRACKS;             // chunks per batch
    const unsigned chunks_per_wav = LANES * NSTEPS;              // 512
    const unsigned waves_x  = (cpb + chunks_per_wav - 1) / chunks_per_wav;  // 480
    const unsigned blocks_x = (waves_x + WAVES_PB - 1) / WAVES_PB;          // 120

    dim3 grid(blocks_x, (unsigned)B, 1);
    rotate_chord_gather_async<<<grid, WAVES_PB * LANES, 0, stream>>>(
        x, lengths, out, B, N, max_len, cpb);
}